// RecognitionLoss_1889785610736
// MI455X (gfx1250) — compile-verified
//
#include <hip/hip_runtime.h>
#include <cstdint>
#include <cstddef>

#define TT 512
#define BB 256
#define CC 128
#define LLAB 64
#define SS 129            // 2*L+1
#define NEGV (-1.0e30f)
#define LOG2E 1.4426950408889634f
#define LN2   0.6931471805599453f

typedef int v4i __attribute__((ext_vector_type(4)));

// ---------------- CDNA5 async global->LDS helpers ----------------
__device__ __forceinline__ void async_copy16(const float* g, float* l) {
#if defined(__has_builtin) && __has_builtin(__builtin_amdgcn_global_load_async_to_lds_b128)
  __builtin_amdgcn_global_load_async_to_lds_b128(
      (__attribute__((address_space(1))) v4i*)(uintptr_t)g,
      (__attribute__((address_space(3))) v4i*)(uint32_t)(uintptr_t)l,
      0, 0);
#else
  asm volatile("global_load_async_to_lds_b128 %0, %1, off"
               :
               : "v"((uint32_t)(uintptr_t)l), "v"(g)
               : "memory");
#endif
}

__device__ __forceinline__ void wait_async_le1() {
  asm volatile("" ::: "memory");
#if defined(__has_builtin) && __has_builtin(__builtin_amdgcn_s_wait_asynccnt)
  __builtin_amdgcn_s_wait_asynccnt(1);
#else
  asm volatile("s_wait_asynccnt 0x1" ::: "memory");
#endif
  asm volatile("" ::: "memory");
}

__device__ __forceinline__ void wait_async_0() {
  asm volatile("" ::: "memory");
#if defined(__has_builtin) && __has_builtin(__builtin_amdgcn_s_wait_asynccnt)
  __builtin_amdgcn_s_wait_asynccnt(0);
#else
  asm volatile("s_wait_asynccnt 0x0" ::: "memory");
#endif
  asm volatile("" ::: "memory");
}

__device__ __forceinline__ float fexp2(float x) { return __builtin_amdgcn_exp2f(x); }
__device__ __forceinline__ float flog2(float x) { return __builtin_amdgcn_logf(x); }

// ---------------- Kernel 1: per-(t,b) log-softmax normalizer (scaled to log2) --------
// norm2[t*B+b] = (max_c + ln(sum_c exp(x - max))) * log2(e); one wave per 128-float row.
__global__ __launch_bounds__(256) void ctc_rownorm(const float* __restrict__ logits,
                                                   float* __restrict__ norm2) {
  const int lane = threadIdx.x & 31;
  const int wave = threadIdx.x >> 5;
  const int row  = blockIdx.x * 8 + wave;               // row in [0, T*B)
  const float4 v = ((const float4*)(logits + (size_t)row * CC))[lane];
  float m = fmaxf(fmaxf(v.x, v.y), fmaxf(v.z, v.w));
#pragma unroll
  for (int off = 16; off >= 1; off >>= 1) m = fmaxf(m, __shfl_xor(m, off, 32));
  float s = __expf(v.x - m) + __expf(v.y - m) + __expf(v.z - m) + __expf(v.w - m);
#pragma unroll
  for (int off = 16; off >= 1; off >>= 1) s += __shfl_xor(s, off, 32);
  if (lane == 0) norm2[row] = (m + __logf(s)) * LOG2E;
}

// ---------------- Kernel 2: CTC forward recursion (log2 domain), one wave/batch ------
__global__ __launch_bounds__(32) void ctc_alpha(const float* __restrict__ logits,
                                                const int* __restrict__ labels,
                                                const int* __restrict__ labels_len,
                                                const int* __restrict__ logits_len,
                                                const float* __restrict__ norm2,
                                                float* __restrict__ out) {
  const int b    = blockIdx.x;
  const int lane = threadIdx.x;                         // 0..31

  __shared__ float rowbuf[3][CC];                       // triple-buffered logits row
  __shared__ float alpha_sh[SS + 3];

  const int len  = labels_len[b];
  const int tlen = logits_len[b];
  const int send = 2 * len;

  // Per-lane state metadata: states s = 5*lane + i
  int   ext[5];
  bool  sval[5];   // s < 2*len+1 and s < S
  bool  skipb[5];  // allowed to take s-2 transition
  bool  valid[5];  // s < S
#pragma unroll
  for (int i = 0; i < 5; ++i) {
    const int s = lane * 5 + i;
    const bool v = (s < SS);
    valid[i] = v;
    int e = 0;
    bool sk = false;
    if (v && (s & 1)) {
      const int j = (s - 1) >> 1;
      e = labels[b * LLAB + j];
      if (s >= 3) {
        const int e2 = labels[b * LLAB + j - 1];
        sk = (e != 0) && (e != e2);
      }
    }
    ext[i]   = e;
    skipb[i] = sk;
    sval[i]  = v && (s < 2 * len + 1);
  }

  // t = 0: alpha2[0] = emit2(0,0), alpha2[1] = emit2(0,1), else NEG (log2 domain)
  const float n20 = norm2[b];
  float a[5];
#pragma unroll
  for (int i = 0; i < 5; ++i) {
    const int s = lane * 5 + i;
    float e = NEGV;
    if (valid[i] && s <= 1)
      e = __builtin_fmaf(logits[(size_t)b * CC + ext[i]], LOG2E, -n20);
    a[i] = e;
  }

  // Prologue: async-prefetch row t=1 into rowbuf[1]
  {
    const float* r1 = logits + ((size_t)1 * BB + b) * CC;
    async_copy16(r1 + 4 * lane, &rowbuf[1][4 * lane]);
  }
  float n2_cur = norm2[(size_t)1 * BB + b];

  int cur = 1;                                          // buffer holding row t
  for (int t = 1; t < TT; ++t) {
    const int nxt = (cur == 2) ? 0 : cur + 1;

    // Prefetch row t+1 (clamped re-fetch on last iter; harmless).
    // Triple buffering: rowbuf[nxt] was last read at iteration t-2 -> no WAR hazard.
    const int tp = (t + 1 < TT) ? (t + 1) : t;
    const float* rp = logits + ((size_t)tp * BB + b) * CC;
    async_copy16(rp + 4 * lane, &rowbuf[nxt][4 * lane]);
    const float n2_nn = norm2[(size_t)tp * BB + b];

    // Ensure rowbuf[cur] has landed (async completions are in-order per wave)
    wait_async_le1();

    // Cross-lane neighbors for s-1 / s-2
    float pa3 = __shfl_up(a[3], 1, 32);
    float pa4 = __shfl_up(a[4], 1, 32);
    if (lane == 0) { pa3 = NEGV; pa4 = NEGV; }

    float na[5];
#pragma unroll
    for (int i = 0; i < 5; ++i) {
      const float e  = sval[i]
                     ? __builtin_fmaf(rowbuf[cur][ext[i]], LOG2E, -n2_cur)
                     : NEGV;
      const float a0 = a[i];
      const float a1 = (i == 0) ? pa4 : a[i - 1];
      float       a2 = (i == 0) ? pa3 : ((i == 1) ? pa4 : a[i - 2]);
      a2 = skipb[i] ? a2 : NEGV;
      const float m   = fmaxf(a0, fmaxf(a1, a2));
      const float sum = fexp2(a0 - m) + fexp2(a1 - m) + fexp2(a2 - m);
      const float nv  = m + flog2(sum) + e;
      na[i] = (t < tlen) ? nv : a0;
    }
#pragma unroll
    for (int i = 0; i < 5; ++i) a[i] = na[i];
    n2_cur = n2_nn;
    cur = nxt;
  }
  wait_async_0();

  // Final: loss = -ln2 * log2addexp2(alpha2[2*len], alpha2[2*len-1])
#pragma unroll
  for (int i = 0; i < 5; ++i)
    if (valid[i]) alpha_sh[lane * 5 + i] = a[i];
  __syncthreads();
  if (lane == 0) {
    const float x = alpha_sh[send];
    const float y = alpha_sh[send - 1];
    const float m = fmaxf(x, y);
    const float ll2 = m + flog2(fexp2(x - m) + fexp2(y - m));
    out[b] = -ll2 * LN2;
  }
}

// ---------------- Host launcher ----------------
extern "C" void kernel_launch(void* const* d_in, const int* in_sizes, int n_in,
                              void* d_out, int out_size, void* d_ws, size_t ws_size,
                              hipStream_t stream) {
  const float* logits     = (const float*)d_in[0];   // (T,B,C) f32
  const int*   labels     = (const int*)d_in[1];     // (B,L) i32
  const int*   labels_len = (const int*)d_in[2];     // (B,)  i32
  const int*   logits_len = (const int*)d_in[3];     // (B,)  i32
  float*       out        = (float*)d_out;           // (B,)  f32
  float*       norm2      = (float*)d_ws;            // T*B floats = 512 KB scratch

  (void)in_sizes; (void)n_in; (void)out_size; (void)ws_size;

  // Pass 1: 131072 rows, 8 rows per 256-thread block (bandwidth-bound, ~3 us roofline)
  ctc_rownorm<<<(TT * BB) / 8, 256, 0, stream>>>(logits, norm2);
  // Pass 2: one wave per batch element, async-pipelined row streaming
  ctc_alpha<<<BB, 32, 0, stream>>>(logits, labels, labels_len, logits_len, norm2, out);
}